// VectorQuantizer_8297876816208
// MI455X (gfx1250) — compile-verified
//
#include <hip/hip_runtime.h>

#define N_ROWS 65536
#define EMB    256
#define NCODE  1024
#define D4     (EMB / 4)        // 64 k-steps of 4 for wmma f32 16x16x4
#define KTILE  32               // codebook rows staged in LDS per tile
#define NTILES (NCODE / KTILE)  // 32
#define LDSS   264              // padded row stride in floats (256 + 8)

typedef float v2f __attribute__((ext_vector_type(2)));
typedef float v8f __attribute__((ext_vector_type(8)));

#if __has_builtin(__builtin_amdgcn_global_load_async_to_lds_b128)
#define USE_ASYNC_LDS 1
#else
#define USE_ASYNC_LDS 0
#endif

#if USE_ASYNC_LDS
typedef int v4i __attribute__((vector_size(16)));
typedef __attribute__((address_space(1))) v4i g_v4i;
typedef __attribute__((address_space(3))) v4i l_v4i;

__device__ __forceinline__ g_v4i* as_global_v4(const void* p) {
  return (g_v4i*)(uintptr_t)p;
}
__device__ __forceinline__ l_v4i* as_lds_v4(const void* p) {
  // generic LDS addresses carry the LDS byte offset in the low 32 bits
  return (l_v4i*)(uint32_t)(uintptr_t)p;
}
__device__ __forceinline__ void wait_asynccnt_le8() {
#if __has_builtin(__builtin_amdgcn_s_wait_asynccnt)
  __builtin_amdgcn_s_wait_asynccnt(8);
#else
  asm volatile("s_wait_asynccnt 0x8" ::: "memory");
#endif
}
__device__ __forceinline__ void wait_asynccnt_le0() {
#if __has_builtin(__builtin_amdgcn_s_wait_asynccnt)
  __builtin_amdgcn_s_wait_asynccnt(0);
#else
  asm volatile("s_wait_asynccnt 0x0" ::: "memory");
#endif
}
#endif

// ---------------- codebook squared norms ----------------
__global__ __launch_bounds__(256) void cnorm_kernel(const float* __restrict__ cb,
                                                    float* __restrict__ cnorm) {
  __shared__ float red[256];
  const int row = blockIdx.x;
  float v = cb[(size_t)row * EMB + threadIdx.x];
  red[threadIdx.x] = v * v;
  __syncthreads();
  for (int s = 128; s > 0; s >>= 1) {
    if (threadIdx.x < s) red[threadIdx.x] += red[threadIdx.x + s];
    __syncthreads();
  }
  if (threadIdx.x == 0) cnorm[row] = red[0];
}

// ---------------- distance + argmin via FP32 WMMA ----------------
// Wave32: each wave owns 16 rows of z. A-fragments (16x4 f32): lane l holds
// z[row = l&15][d0 + 2*(l>>4) + {0,1}]. B-fragments (4x16): lane l holds
// cb[code = kbase + (l&15)][d0 + 2*(l>>4) + {0,1}] (B = C^T => row-major cb).
// C/D 16x16 f32: vgpr r, lane l -> D[r + 8*(l>>4)][l&15].
__global__ __launch_bounds__(256) void vq_argmin_kernel(
    const float* __restrict__ z, const float* __restrict__ cb,
    const float* __restrict__ cnorm, float* __restrict__ codes_f) {
#if USE_ASYNC_LDS
  __shared__ float smem[2][KTILE * LDSS];
#else
  __shared__ float smem[1][KTILE * LDSS];
#endif

  const int t    = threadIdx.x;
  const int wave = t >> 5;
  const int lane = t & 31;
  const int col  = lane & 15;
  const int half = lane >> 4;     // 0 or 1
  const int koff = half * 2;
  const int rowbase = (blockIdx.x * 8 + wave) * 16;

  // Load this wave's 16 z-rows fully into A-fragment registers (128 VGPRs).
  v2f a[D4];
  const float* zrow = z + (size_t)(rowbase + col) * EMB;
#pragma unroll
  for (int dc = 0; dc < D4; ++dc)
    a[dc] = *(const v2f*)(zrow + dc * 4 + koff);

  float best[8];
  int   bidx[8];
#pragma unroll
  for (int r = 0; r < 8; ++r) { best[r] = 3.4e38f; bidx[r] = 0; }

#if USE_ASYNC_LDS
  // Prologue: async-stage tile 0 into buffer 0. 8 b128 per thread => 8 async
  // instructions per wave, tracked in-order on ASYNCcnt.
#pragma unroll
  for (int i = 0; i < 8; ++i) {
    const int idx = i * 256 + t;
    const int r  = idx >> 6;
    const int c4 = idx & 63;
    __builtin_amdgcn_global_load_async_to_lds_b128(
        as_global_v4(cb + (size_t)r * EMB + c4 * 4),
        as_lds_v4(&smem[0][r * LDSS + c4 * 4]), 0, 0);
  }
#endif

  for (int tile = 0; tile < NTILES; ++tile) {
    const int kbase0 = tile * KTILE;
#if USE_ASYNC_LDS
    const int buf = tile & 1;
    __syncthreads();  // all waves done reading buf^1 from the previous round
    if (tile + 1 < NTILES) {
#pragma unroll
      for (int i = 0; i < 8; ++i) {
        const int idx = i * 256 + t;
        const int r  = idx >> 6;
        const int c4 = idx & 63;
        __builtin_amdgcn_global_load_async_to_lds_b128(
            as_global_v4(cb + (size_t)(kbase0 + KTILE + r) * EMB + c4 * 4),
            as_lds_v4(&smem[buf ^ 1][r * LDSS + c4 * 4]), 0, 0);
      }
      wait_asynccnt_le8();  // oldest 8 (this tile's loads) have landed
    } else {
      wait_asynccnt_le0();
    }
    __syncthreads();        // every wave's loads for this tile have landed
    const float* tileptr = smem[buf];
#else
    __syncthreads();  // protect LDS from previous iteration's readers
#pragma unroll
    for (int i = 0; i < (KTILE * (EMB / 4)) / 256; ++i) {
      const int idx = i * 256 + t;
      const int r  = idx >> 6;
      const int c4 = idx & 63;
      const float4 val = ((const float4*)cb)[(size_t)(kbase0 + r) * (EMB / 4) + c4];
      *(float4*)&smem[0][r * LDSS + c4 * 4] = val;
    }
    __syncthreads();
    const float* tileptr = smem[0];
#endif

    for (int kc = 0; kc < KTILE / 16; ++kc) {
      const int kbase = kbase0 + kc * 16;
      const float* brow = &tileptr[(kc * 16 + col) * LDSS + koff];

      // Two independent accumulator chains for XDL pipelining.
      v8f acc0 = {}, acc1 = {};
#pragma unroll
      for (int dc = 0; dc < D4; dc += 2) {
        v2f b0 = *(const v2f*)(brow + dc * 4);
        v2f b1 = *(const v2f*)(brow + dc * 4 + 4);
        acc0 = __builtin_amdgcn_wmma_f32_16x16x4_f32(
            false, a[dc],     false, b0, (short)0, acc0, false, false);
        acc1 = __builtin_amdgcn_wmma_f32_16x16x4_f32(
            false, a[dc + 1], false, b1, (short)0, acc1, false, false);
      }

      const float cn  = cnorm[kbase + col];
      const int  code = kbase + col;
#pragma unroll
      for (int r = 0; r < 8; ++r) {
        const float sc = cn - 2.0f * (acc0[r] + acc1[r]);  // ||c||^2 - 2 z.c
        if (sc < best[r]) { best[r] = sc; bidx[r] = code; }
      }
    }
  }

  // Min-reduce across the 16 lanes of each half (each lane = distinct code col).
#pragma unroll
  for (int r = 0; r < 8; ++r) {
    float v  = best[r];
    int   id = bidx[r];
#pragma unroll
    for (int m = 8; m >= 1; m >>= 1) {
      const float ov = __shfl_xor(v, m, 16);
      const int   oi = __shfl_xor(id, m, 16);
      if (ov < v || (ov == v && oi < id)) { v = ov; id = oi; }
    }
    if (col == 0) codes_f[rowbase + half * 8 + r] = (float)id;
  }
}

// ---------------- gather + STE output + loss partial sums ----------------
__global__ __launch_bounds__(256) void vq_gather_loss_kernel(
    const float* __restrict__ z, const float* __restrict__ cb,
    const float* __restrict__ codes_f, float* __restrict__ out,
    float* __restrict__ acc) {
  const int t   = threadIdx.x;
  const int row = blockIdx.x * 4 + (t >> 6);
  const int c4  = t & 63;

  const int code = (int)codes_f[row];
  const float4 q  = ((const float4*)cb)[(size_t)code * (EMB / 4) + c4];
  const float4 zz = ((const float4*)z)[(size_t)row * (EMB / 4) + c4];

  const float d0 = q.x - zz.x, d1 = q.y - zz.y, d2 = q.z - zz.z, d3 = q.w - zz.w;
  float4 o;  // straight-through: z + (q - z), same op order as reference
  o.x = zz.x + d0; o.y = zz.y + d1; o.z = zz.z + d2; o.w = zz.w + d3;
  ((float4*)out)[(size_t)row * (EMB / 4) + c4] = o;

  float s = d0 * d0 + d1 * d1 + d2 * d2 + d3 * d3;

  __shared__ float red[256];
  red[t] = s;
  __syncthreads();
  for (int st = 128; st > 0; st >>= 1) {
    if (t < st) red[t] += red[t + st];
    __syncthreads();
  }
  if (t == 0) unsafeAtomicAdd(acc, red[0]);
}

__global__ void zero_acc_kernel(float* acc) { acc[0] = 0.0f; }

__global__ void finalize_kernel(const float* __restrict__ acc,
                                float* __restrict__ loss_out) {
  // cb_loss == cmt_loss numerically; vq_loss = (1 + BETA) * mean((q-z)^2)
  loss_out[0] = 1.25f * acc[0] / 16777216.0f;   // N_ROWS * EMB
}

extern "C" void kernel_launch(void* const* d_in, const int* in_sizes, int n_in,
                              void* d_out, int out_size, void* d_ws, size_t ws_size,
                              hipStream_t stream) {
  const float* z  = (const float*)d_in[0];   // [65536, 256]
  const float* cb = (const float*)d_in[1];   // [1024, 256]

  float* out     = (float*)d_out;                    // [65536*256]
  float* codes_f = out + (size_t)N_ROWS * EMB;       // [65536] (codes as float)
  float* loss    = codes_f + N_ROWS;                 // [1]

  float* cnorm = (float*)d_ws;                       // [1024]
  float* acc   = cnorm + NCODE;                      // [1]

  zero_acc_kernel<<<1, 1, 0, stream>>>(acc);
  cnorm_kernel<<<NCODE, 256, 0, stream>>>(cb, cnorm);
  vq_argmin_kernel<<<N_ROWS / 128, 256, 0, stream>>>(z, cb, cnorm, codes_f);
  vq_gather_loss_kernel<<<N_ROWS / 4, 256, 0, stream>>>(z, cb, codes_f, out, acc);
  finalize_kernel<<<1, 1, 0, stream>>>(acc, loss);
}